// SobelLayer_7395933683874
// MI455X (gfx1250) — compile-verified
//
#include <hip/hip_runtime.h>
#include <cstdint>
#include <cstddef>

// ---------------------------------------------------------------------------
// Sobel depthwise 3x3 + magnitude + channel-mean, NCHW f32.
//   x: [B, 64, 256, 256] f32  ->  out: [B, 1, 256, 256] f32
//
// Memory-bound: 268 MB in / 4 MB out  =>  ~11.7 us floor at 23.3 TB/s HBM.
// Each block owns a 16x64 spatial tile and loops the 64 channels,
// double-buffering an 18x66 halo tile in LDS via gfx1250 async global->LDS
// loads (ASYNCcnt) so HBM streaming overlaps the f32 stencil math.
// No WMMA: there is no matrix contraction in this op (per-pixel private
// channel accumulation), and f32 precision must be preserved.
// ---------------------------------------------------------------------------

#define TW 64
#define TH 16
#define TCOLS (TW + 2)               // 66
#define TROWS (TH + 2)               // 18
#define PITCH 68                     // LDS floats per tile row (padded)
#define TILE_ELEMS (TROWS * TCOLS)   // 1188
#define BUF_FLOATS (TROWS * PITCH)   // 1224
#define NISSUE 5                     // ceil(1188/256); every lane issues all 5
                                     // so per-wave ASYNCcnt is uniform

__global__ __launch_bounds__(256)
void sobel_mean_kernel(const float* __restrict__ x, float* __restrict__ out) {
  __shared__ float tile[2 * BUF_FLOATS + 1];   // last word = dummy async sink

  const int tid = threadIdx.x;
  const int w0 = blockIdx.x * TW;
  const int h0 = blockIdx.y * TH;
  const int b  = blockIdx.z;

  // Low 32 bits of a flat pointer into __shared__ == LDS byte address.
  const uint32_t lds_base  = (uint32_t)(uintptr_t)&tile[0];
  const uint32_t lds_dummy = lds_base + (uint32_t)(2 * BUF_FLOATS) * 4u;

  // Channel-invariant per-slot info: LDS target, global byte offset (clamped
  // to a safe in-plane address), and which LDS slot needs an explicit zero
  // (zero-padding at image borders).
  uint32_t lds_rel[NISSUE];
  int32_t  gbyte[NISSUE];
  int      zslot[NISSUE];
#pragma unroll
  for (int i = 0; i < NISSUE; ++i) {
    const int idx = tid + i * 256;               // 0..1279
    const bool slot_ok = idx < TILE_ELEMS;       // real tile element?
    const int row = idx / TCOLS;
    const int col = idx - row * TCOLS;
    const int gh = h0 - 1 + row;
    const int gw = w0 - 1 + col;
    const bool img_ok =
        slot_ok && (gh >= 0) && (gh < 256) && (gw >= 0) && (gw < 256);
    const int ghc = gh < 0 ? 0 : (gh > 255 ? 255 : gh);
    const int gwc = gw < 0 ? 0 : (gw > 255 ? 255 : gw);
    gbyte[i]   = (ghc * 256 + gwc) * 4;          // always a valid address
    lds_rel[i] = img_ok ? (uint32_t)(row * PITCH + col) * 4u : 0xFFFFFFFFu;
    zslot[i]   = (slot_ok && !img_ok) ? (row * PITCH + col) : -1;
  }

  const float* plane0 = x + (size_t)b * 64u * 256u * 256u;

  // Issue exactly NISSUE async global->LDS loads per lane for channel c into
  // buffer `buf`. OOB lanes are redirected to the dummy LDS word so ASYNCcnt
  // advances identically on every wave (wait<=5 then provably drains the
  // previous buffer, since async loads complete in order).
  auto issue = [&](int c, int buf) {
    const uint64_t base =
        (uint64_t)(uintptr_t)(plane0 + (size_t)c * 256u * 256u);
    const uint32_t bufbase =
        lds_base + (uint32_t)buf * (uint32_t)(BUF_FLOATS * 4);
#pragma unroll
    for (int i = 0; i < NISSUE; ++i) {
      const uint32_t la =
          (lds_rel[i] == 0xFFFFFFFFu) ? lds_dummy : (bufbase + lds_rel[i]);
      asm volatile("global_load_async_to_lds_b32 %0, %1, %2"
                   :
                   : "v"(la), "v"(gbyte[i]), "s"(base)
                   : "memory");
      if (zslot[i] >= 0) tile[buf * BUF_FLOATS + zslot[i]] = 0.0f;  // zero-pad
    }
  };

  float acc[4] = {0.f, 0.f, 0.f, 0.f};
  const int col = tid & (TW - 1);   // 0..63
  const int r0  = (tid >> 6) * 4;   // 0,4,8,12 : 4 contiguous output rows

  issue(0, 0);
  for (int c = 0; c < 64; ++c) {
    const int cur = c & 1;
    if (c + 1 < 64) {
      issue(c + 1, cur ^ 1);                       // prefetch next channel
      asm volatile("s_wait_asynccnt 5" ::: "memory");  // drain current buffer
    } else {
      asm volatile("s_wait_asynccnt 0" ::: "memory");
    }
    __syncthreads();  // cross-wave: everyone's buffer writes visible

    const float* t = &tile[cur * BUF_FLOATS];
    // Row-sum sharing: 6 halo rows feed 4 output pixels.
    float d[6], s[6];
#pragma unroll
    for (int j = 0; j < 6; ++j) {
      const float* rp = t + (r0 + j) * PITCH + col;
      const float a = rp[0], m = rp[1], e = rp[2];
      d[j] = e - a;                 // horizontal difference
      s[j] = a + 2.0f * m + e;      // horizontal smoothing
    }
#pragma unroll
    for (int k = 0; k < 4; ++k) {
      const float ex = d[k] + 2.0f * d[k + 1] + d[k + 2];
      const float ey = s[k + 2] - s[k];
      acc[k] += sqrtf(ex * ex + ey * ey + 1e-12f);
    }
    __syncthreads();  // protect this buffer from next iteration's prefetch
  }

  float* op = out + ((size_t)b * 256 + (size_t)(h0 + r0)) * 256 + (w0 + col);
#pragma unroll
  for (int k = 0; k < 4; ++k)
    op[(size_t)k * 256] = acc[k] * 0.015625f;   // 1/64 channel mean
}

extern "C" void kernel_launch(void* const* d_in, const int* in_sizes, int n_in,
                              void* d_out, int out_size, void* d_ws,
                              size_t ws_size, hipStream_t stream) {
  (void)n_in; (void)d_ws; (void)ws_size; (void)out_size;
  const float* x = (const float*)d_in[0];
  float* out = (float*)d_out;
  const int B = in_sizes[0] / (64 * 256 * 256);   // 16
  dim3 grid(256 / TW, 256 / TH, B);               // 4 x 16 x B
  dim3 block(256);
  hipLaunchKernelGGL(sobel_mean_kernel, grid, block, 0, stream, x, out);
}